// CondensateAndSum_59030030516972
// MI455X (gfx1250) — compile-verified
//
#include <hip/hip_runtime.h>
#include <stdint.h>

#define RADIUS2   0.64f
#define MIN_BETA  0.1f
#define MAX_ITERS 100
#define NFEAT     64

typedef float v2f __attribute__((ext_vector_type(2)));
typedef float v8f __attribute__((ext_vector_type(8)));

// ---------------- init: zero summed, asso=-1, zero argmax slots ----------------
__global__ void cs_init_kernel(float* __restrict__ summed, int* __restrict__ asso,
                               unsigned long long* __restrict__ slots, int N) {
    long long total = (long long)N * NFEAT;
    long long stride = (long long)gridDim.x * blockDim.x;
    for (long long t = (long long)blockIdx.x * blockDim.x + threadIdx.x; t < total; t += stride) {
        summed[t] = 0.0f;
        if (t < N)         asso[t]  = -1;
        if (t < MAX_ITERS) slots[t] = 0ULL;
    }
}

// ---------------- per-iteration argmax of beta among active points ----------------
__global__ void cs_argmax_kernel(const float* __restrict__ betas,
                                 const int* __restrict__ asso,
                                 unsigned long long* __restrict__ slot, int N) {
    unsigned long long key = 0ULL;
    for (int j = blockIdx.x * blockDim.x + threadIdx.x; j < N; j += gridDim.x * blockDim.x) {
        float b = betas[j];
        if (asso[j] < 0 && b >= MIN_BETA) {
            unsigned long long k =
                ((unsigned long long)__float_as_uint(b) << 32) | (unsigned int)(~j);
            if (k > key) key = k;
        }
    }
    // wave32 reduction
    for (int off = 16; off > 0; off >>= 1) {
        unsigned long long o = __shfl_xor(key, off, 32);
        if (o > key) key = o;
    }
    __shared__ unsigned long long red[8];
    int lane = threadIdx.x & 31;
    int wid  = threadIdx.x >> 5;
    if (lane == 0) red[wid] = key;
    __syncthreads();
    if (wid == 0) {
        int nw = blockDim.x >> 5;
        unsigned long long v = (lane < nw) ? red[lane] : 0ULL;
        for (int off = 16; off > 0; off >>= 1) {
            unsigned long long o = __shfl_xor(v, off, 32);
            if (o > v) v = o;
        }
        if (lane == 0 && v != 0ULL) atomicMax(slot, v);
    }
}

// ---------------- per-iteration gather: radius test + WMMA feature sum ----------------
__global__ void cs_scatter_kernel(const float* __restrict__ ccoords,
                                  const float* __restrict__ features,
                                  const int* __restrict__ row_splits, int nsplits,
                                  float* __restrict__ summed, int* __restrict__ asso,
                                  const unsigned long long* __restrict__ slot, int N) {
    unsigned long long key = *slot;
    if (key == 0ULL) return;                    // no active point this iteration
    int i = (int)(~(unsigned int)key);          // winner index (lowest-index tie-break)

    float xi = ccoords[2 * i], yi = ccoords[2 * i + 1];
    // event interval of i (seg equality <=> same [lo,hi) interval)
    int lo = 0, hi = N;
    for (int e = 0; e + 1 < nsplits; ++e) {
        int a = row_splits[e], b = row_splits[e + 1];
        if (i >= a && i < b) { lo = a; hi = b; }
    }

    __shared__ float part[NFEAT];
    __shared__ int   anySel;
    if (threadIdx.x == 0) anySel = 0;
    if (threadIdx.x < NFEAT) part[threadIdx.x] = 0.0f;
    __syncthreads();

    int lane   = threadIdx.x & 31;
    int wid    = (blockIdx.x * blockDim.x + threadIdx.x) >> 5;
    int nwaves = (gridDim.x * blockDim.x) >> 5;

    v8f acc0 = {}, acc1 = {}, acc2 = {}, acc3 = {};
    bool touched = false;

    for (int base = lo + wid * 32; base < hi; base += nwaves * 32) {
        int  j   = base + lane;
        bool inr = j < hi;
        int  jc  = inr ? j : (hi - 1);
        float dx = ccoords[2 * jc]     - xi;
        float dy = ccoords[2 * jc + 1] - yi;
        bool sel = inr && (asso[jc] < 0) && (dx * dx + dy * dy <= RADIUS2);
        unsigned int bal = __builtin_amdgcn_ballot_w32(sel);
        if (sel) asso[j] = i;
        if (bal == 0u) continue;                 // wave-uniform skip
        touched = true;

        for (int g = 0; g < 8; ++g) {            // 8 groups of 4 rows
            unsigned int grp = (bal >> (4 * g)) & 0xFu;      // wave-uniform (SGPR)
            if (grp == 0u) continue;                         // wave-uniform skip
            int j0 = base + 4 * g;
            // masks come straight from the (scalar) ballot bits -- no shuffles needed
            float m0 = (grp & 1u) ? 1.0f : 0.0f;
            float m1 = (grp & 2u) ? 1.0f : 0.0f;
            float m2 = (grp & 4u) ? 1.0f : 0.0f;
            float m3 = (grp & 8u) ? 1.0f : 0.0f;
            // A (16x4 f32): only row M=0 nonzero -> lane0 holds K0,K1; lane16 holds K2,K3
            v2f A;
            A.x = (lane == 0) ? m0 : (lane == 16) ? m2 : 0.0f;
            A.y = (lane == 0) ? m1 : (lane == 16) ? m3 : 0.0f;
            // B (4x16 f32): VGPR0 = rows K0(lanes0-15)/K2(lanes16-31), VGPR1 = K1/K3
            int rA = (lane < 16) ? j0 : j0 + 2;
            int rB = rA + 1;
            rA = rA < hi ? rA : hi - 1;          // clamp; masked rows contribute 0 via A
            rB = rB < hi ? rB : hi - 1;
            int cb = lane & 15;
            const float* f = features;
            v2f B0, B1, B2, B3;
            B0.x = f[rA * 64 +  0 + cb];  B0.y = f[rB * 64 +  0 + cb];
            B1.x = f[rA * 64 + 16 + cb];  B1.y = f[rB * 64 + 16 + cb];
            B2.x = f[rA * 64 + 32 + cb];  B2.y = f[rB * 64 + 32 + cb];
            B3.x = f[rA * 64 + 48 + cb];  B3.y = f[rB * 64 + 48 + cb];
            acc0 = __builtin_amdgcn_wmma_f32_16x16x4_f32(false, A, false, B0, (short)0, acc0, false, false);
            acc1 = __builtin_amdgcn_wmma_f32_16x16x4_f32(false, A, false, B1, (short)0, acc1, false, false);
            acc2 = __builtin_amdgcn_wmma_f32_16x16x4_f32(false, A, false, B2, (short)0, acc2, false, false);
            acc3 = __builtin_amdgcn_wmma_f32_16x16x4_f32(false, A, false, B3, (short)0, acc3, false, false);
        }
    }

    // D row M=0 lives in element 0, lanes 0..15
    if (touched) {
        if (lane < 16) {
            atomicAdd(&part[ 0 + lane], acc0[0]);
            atomicAdd(&part[16 + lane], acc1[0]);
            atomicAdd(&part[32 + lane], acc2[0]);
            atomicAdd(&part[48 + lane], acc3[0]);
        }
        anySel = 1;
    }
    __syncthreads();
    if (anySel != 0 && threadIdx.x < NFEAT) {
        atomicAdd(&summed[i * NFEAT + threadIdx.x], part[threadIdx.x]);
    }
}

// ---------------- launcher ----------------
extern "C" void kernel_launch(void* const* d_in, const int* in_sizes, int n_in,
                              void* d_out, int out_size, void* d_ws, size_t ws_size,
                              hipStream_t stream) {
    const float* ccoords    = (const float*)d_in[0];
    const float* betas      = (const float*)d_in[1];
    const float* features   = (const float*)d_in[2];
    const int*   row_splits = (const int*)d_in[3];
    int N       = in_sizes[1];      // betas is (N,1)
    int nsplits = in_sizes[3];      // E+1

    float* summed = (float*)d_out;
    int*   asso   = (int*)((float*)d_out + (size_t)N * NFEAT);
    unsigned long long* slots = (unsigned long long*)d_ws;

    cs_init_kernel<<<2048, 256, 0, stream>>>(summed, asso, slots, N);
    for (int k = 0; k < MAX_ITERS; ++k) {
        cs_argmax_kernel<<<256, 256, 0, stream>>>(betas, asso, slots + k, N);
        cs_scatter_kernel<<<256, 256, 0, stream>>>(ccoords, features, row_splits, nsplits,
                                                   summed, asso, slots + k, N);
    }
}